// VQVAE_85847806313146
// MI455X (gfx1250) — compile-verified
//
#include <hip/hip_runtime.h>
#include <hip/hip_bf16.h>

// ---------------------------------------------------------------------------
// VQ-VAE quantize + EMA update for MI455X (gfx1250, wave32, WMMA).
//
// Shapes: z (64,8,64,64) f32, codebook (512,8) f32, ema_cs (512), ema_w (512,8)
// M = 64*4096 = 262144 vectors, D = 8, K = 512.
//
// Stage 1: per-wave 16-vector tile, distances via V_WMMA_F32_16X16X4_F32
//          (two chained K=4 WMMAs -> 16x16 dot tile). Argmin over 512 codes:
//          cheap (float,int) tracking in the hot loop, then a branchless
//          packed-u64-key v_min_u64 butterfly across lanes (exact lowest-
//          index tie-break). z_q gather, loss partials, atomic counts/sums.
// Stage 2: EMA + Laplace smoothing + loss reduction, pack tuple outputs.
// ---------------------------------------------------------------------------

typedef __attribute__((ext_vector_type(2))) float v2f;
typedef __attribute__((ext_vector_type(8))) float v8f;

#define VQ_K      512
#define VQ_D      8
#define VQ_M      262144          // 64 * 4096
#define VQ_NTILES 16384           // VQ_M / 16
#define VQ_HW     4096
#define VQ_BSTRIDE 32768          // 8 * 4096 floats per batch image
#define DECAY     0.99f
#define ONE_M_DECAY 0.01f
#define VQ_EPS    1e-5f

// output packing offsets (floats)
#define OFF_ZQ   0
#define OFF_LOSS 2097152
#define OFF_CB   2097153
#define OFF_CS   (OFF_CB + VQ_K * VQ_D)     // 2101249
#define OFF_W    (OFF_CS + VQ_K)            // 2101761

__device__ __forceinline__ unsigned int mono_f32(float f) {
    unsigned int b = __float_as_uint(f);
    // order-preserving map float -> uint32
    return (b & 0x80000000u) ? ~b : (b | 0x80000000u);
}

__device__ __forceinline__ unsigned long long shfl_xor_u64(unsigned long long v, int mask) {
    unsigned int lo = (unsigned int)v;
    unsigned int hi = (unsigned int)(v >> 32);
    lo = __shfl_xor(lo, mask, 32);
    hi = __shfl_xor(hi, mask, 32);
    return ((unsigned long long)hi << 32) | lo;
}

__global__ void vq_zero_ws(float* ws, int n) {
    int i = blockIdx.x * blockDim.x + threadIdx.x;
    if (i < n) ws[i] = 0.0f;
}

// 256 threads = 8 waves per block; each wave handles one 16-vector M tile.
__global__ __launch_bounds__(256)
void vq_stage1(const float* __restrict__ z,
               const float* __restrict__ codebook,
               float* __restrict__ out_zq,
               float* __restrict__ counts,
               float* __restrict__ sums,
               float* __restrict__ lossPartials) {
    __shared__ float lds_cb[VQ_K * VQ_D];   // 16 KB
    __shared__ float lds_esq[VQ_K];         // 2 KB

    // cooperative codebook + ||e||^2 load
    for (int k = threadIdx.x; k < VQ_K; k += 256) {
        float s = 0.0f;
#pragma unroll
        for (int c = 0; c < VQ_D; ++c) {
            float v = codebook[k * VQ_D + c];
            lds_cb[k * VQ_D + c] = v;
            s += v * v;
        }
        lds_esq[k] = s;
    }
    __syncthreads();

    const int lane   = threadIdx.x & 31;
    const int waveId = threadIdx.x >> 5;
    const int tileM  = blockIdx.x * 8 + waveId;   // 0 .. 16383

    // ---- A fragment: 16x8 slice of flat latents (two 16x4 K-chunks) ----
    // A layout (32-bit 16x4): lane L holds row (L&15); VGPR0=K c0, VGPR1=K c0+1
    // with c0 = 0 for lanes 0-15, c0 = 2 for lanes 16-31.
    const int rowA = lane & 15;
    const int mA   = tileM * 16 + rowA;
    const int bA   = mA >> 12;
    const int nA   = mA & (VQ_HW - 1);
    const float* zr = z + (size_t)bA * VQ_BSTRIDE + nA;
    const int c0 = (lane & 16) ? 2 : 0;

    v2f a_lo, a_hi;
    a_lo.x = zr[(size_t)(c0 + 0) * VQ_HW];
    a_lo.y = zr[(size_t)(c0 + 1) * VQ_HW];
    a_hi.x = zr[(size_t)(c0 + 4) * VQ_HW];
    a_hi.y = zr[(size_t)(c0 + 5) * VQ_HW];

    // running argmin per output row: distance (minus per-row const ||z||^2)
    // and code index; strict < with ascending cn keeps lowest index per lane
    float bestd[8];
    int   bestk[8];
#pragma unroll
    for (int r = 0; r < 8; ++r) { bestd[r] = 3.4e38f; bestk[r] = 0; }

    // ---- sweep 32 N-tiles of 16 codebook entries each ----
    for (int t = 0; t < 32; ++t) {
        const int cn = t * 16 + (lane & 15);      // this lane's column / code idx
        // B layout (32-bit 4x16): lane L holds column (L&15); VGPR0=K c0, VGPR1=K c0+1
        const v2f b_lo = *(const v2f*)&lds_cb[cn * VQ_D + c0];
        const v2f b_hi = *(const v2f*)&lds_cb[cn * VQ_D + c0 + 4];

        v8f acc = {};
        acc = __builtin_amdgcn_wmma_f32_16x16x4_f32(false, a_lo, false, b_lo,
                                                    (short)0, acc, false, false);
        acc = __builtin_amdgcn_wmma_f32_16x16x4_f32(false, a_hi, false, b_hi,
                                                    (short)0, acc, false, false);

        const float esq = lds_esq[cn];
#pragma unroll
        for (int r = 0; r < 8; ++r) {
            // distance^2 minus the per-row constant ||z||^2 (argmin-invariant)
            float dist = __builtin_fmaf(-2.0f, acc[r], esq);
            if (dist < bestd[r]) { bestd[r] = dist; bestk[r] = cn; }
        }
    }

    // ---- pack (dist, idx) into u64 keys, branchless v_min_u64 butterfly
    //      within each 16-lane half (rows 0-7 | rows 8-15); equal dists give
    //      equal high bits so min picks lowest index (matches jnp.argmin) ----
    unsigned long long key[8];
#pragma unroll
    for (int r = 0; r < 8; ++r) {
        key[r] = ((unsigned long long)mono_f32(bestd[r]) << 32) |
                 (unsigned int)bestk[r];
    }
#pragma unroll
    for (int r = 0; r < 8; ++r) {
#pragma unroll
        for (int m = 8; m >= 1; m >>= 1) {
            unsigned long long o = shfl_xor_u64(key[r], m);
            key[r] = (o < key[r]) ? o : key[r];
        }
    }

    // ---- epilogue: rows r and 8+r handled by lanes r and 16+r ----
    float localLoss = 0.0f;
#pragma unroll
    for (int r = 0; r < 8; ++r) {
        if ((lane & 15) == r) {
            const int row  = r + ((lane & 16) ? 8 : 0);
            const int kidx = (int)(key[r] & 0xFFFFFFFFu);
            const int m    = tileM * 16 + row;
            const int b    = m >> 12;
            const int n    = m & (VQ_HW - 1);
            const float* zv = z + (size_t)b * VQ_BSTRIDE + n;
            float* zq = out_zq + (size_t)b * VQ_BSTRIDE + n;
            atomicAdd(&counts[kidx], 1.0f);
#pragma unroll
            for (int c = 0; c < VQ_D; ++c) {
                float e  = lds_cb[kidx * VQ_D + c];
                float zc = zv[(size_t)c * VQ_HW];
                zq[(size_t)c * VQ_HW] = e;
                float d = e - zc;
                localLoss += d * d;
                atomicAdd(&sums[kidx * VQ_D + c], zc);
            }
        }
    }

    // deterministic per-tile loss partial
#pragma unroll
    for (int m = 16; m >= 1; m >>= 1) localLoss += __shfl_xor(localLoss, m, 32);
    if (lane == 0) lossPartials[tileM] = localLoss;
}

// single block, 512 threads: EMA update + smoothing + loss reduction + pack
__global__ __launch_bounds__(512)
void vq_finalize(const float* __restrict__ ema_cs,
                 const float* __restrict__ ema_w,
                 const float* __restrict__ counts,
                 const float* __restrict__ sums,
                 const float* __restrict__ lossPartials,
                 float* __restrict__ out) {
    __shared__ float red[VQ_K];
    const int k = threadIdx.x;

    const float ncs = ema_cs[k] * DECAY + counts[k] * ONE_M_DECAY;
    red[k] = ncs;
    __syncthreads();
#pragma unroll
    for (int s = 256; s >= 1; s >>= 1) {
        if (k < s) red[k] += red[k + s];
        __syncthreads();
    }
    const float ntot = red[0];
    __syncthreads();

    const float cs = (ncs + VQ_EPS) / (ntot + (float)VQ_K * VQ_EPS) * ntot;
    out[OFF_CS + k] = cs;
#pragma unroll
    for (int c = 0; c < VQ_D; ++c) {
        float nw = ema_w[k * VQ_D + c] * DECAY + sums[k * VQ_D + c] * ONE_M_DECAY;
        out[OFF_W + k * VQ_D + c]  = nw;
        out[OFF_CB + k * VQ_D + c] = nw / cs;
    }

    // loss = sum(partials) / numel(z)
    float ls = 0.0f;
    for (int i = k; i < VQ_NTILES; i += VQ_K) ls += lossPartials[i];
    red[k] = ls;
    __syncthreads();
#pragma unroll
    for (int s = 256; s >= 1; s >>= 1) {
        if (k < s) red[k] += red[k + s];
        __syncthreads();
    }
    if (k == 0) out[OFF_LOSS] = red[0] / 2097152.0f;
}

extern "C" void kernel_launch(void* const* d_in, const int* in_sizes, int n_in,
                              void* d_out, int out_size, void* d_ws, size_t ws_size,
                              hipStream_t stream) {
    (void)in_sizes; (void)n_in; (void)out_size; (void)ws_size;
    const float* z        = (const float*)d_in[0];
    const float* codebook = (const float*)d_in[1];
    const float* ema_cs   = (const float*)d_in[2];
    const float* ema_w    = (const float*)d_in[3];
    float* out = (float*)d_out;

    float* counts    = (float*)d_ws;          // 512
    float* sums      = counts + VQ_K;         // 512*8
    float* lossParts = sums + VQ_K * VQ_D;    // 16384

    const int zeroN = VQ_K + VQ_K * VQ_D;     // counts + sums
    vq_zero_ws<<<(zeroN + 255) / 256, 256, 0, stream>>>(counts, zeroN);

    vq_stage1<<<VQ_NTILES / 8, 256, 0, stream>>>(z, codebook,
                                                 out + OFF_ZQ,
                                                 counts, sums, lossParts);

    vq_finalize<<<1, VQ_K, 0, stream>>>(ema_cs, ema_w, counts, sums,
                                        lossParts, out);
}